// SimpleGraphNetwork_60404420051428
// MI455X (gfx1250) — compile-verified
//
#include <hip/hip_runtime.h>
#include <hip/hip_bf16.h>

typedef __attribute__((ext_vector_type(2))) float v2f;
typedef __attribute__((ext_vector_type(8))) float v8f;

#define NEG_SLOPE 0.2f

// ---------------------------------------------------------------------------
// GEMM: C[M x Nc] = A[M x K] * B[K x Nc], fp32 via V_WMMA_F32_16X16X4_F32.
// Block = 4 waves = 64(M) x 64(N) tile. Wave = 16(M) x 64(N) = 4 C-tiles.
// B panel [K x 64] staged in LDS via GLOBAL_LOAD_ASYNC_TO_LDS_B128 (ASYNCcnt
// path, inline asm), shared by all 4 waves. A fragment: one b64 load per
// K-step, reused by 4 WMMAs (one per N-subtile).
// A-operand layout (ISA 7.12.2, 32-bit A 16x4): lanes 0-15 -> M=0..15,K={0,1};
// lanes 16-31 -> K={2,3}. B 4x16: rows striped across lanes (same K split).
// C/D: VGPR v, lanes 0-15 -> (M=v, N=lane&15); lanes 16-31 -> M=8+v.
// ---------------------------------------------------------------------------
#define WMMA_STEP(acc, boff)                                                   \
  do {                                                                         \
    v2f b;                                                                     \
    b.x = bk0[(boff) + nr];                                                    \
    b.y = bk1[(boff) + nr];                                                    \
    acc = __builtin_amdgcn_wmma_f32_16x16x4_f32(false, a, false, b, (short)0,  \
                                                acc, false, false);            \
  } while (0)

__global__ __launch_bounds__(128)
void gemm_wmma_f32(const float* __restrict__ A, const float* __restrict__ B,
                   float* __restrict__ C, int M, int K, int Nc) {
  extern __shared__ float ldsB[];  // [K][64]
  const int wave = threadIdx.x >> 5;
  const int lane = threadIdx.x & 31;
  const int col0 = blockIdx.y * 64;

  // ---- stage B panel [K x 64] into LDS via async global->LDS DMA ----
  const int nvec = (K * 64) >> 2;  // float4 chunks
  for (int c = threadIdx.x; c < nvec; c += 128) {
    const int k = c >> 4;           // 16 float4 per 64-wide row
    const int f = (c & 15) << 2;
    const float* gsrc = B + (size_t)k * Nc + col0 + f;
    // VDST = LDS byte offset (low 32 bits of generic LDS address),
    // VADDR = 64-bit global address pair, SADDR = off.
    unsigned lds_off = (unsigned)(uintptr_t)(&ldsB[(k << 6) + f]);
    unsigned long long gaddr = (unsigned long long)(uintptr_t)gsrc;
    asm volatile("global_load_async_to_lds_b128 %0, %1, off"
                 :
                 : "v"(lds_off), "v"(gaddr)
                 : "memory");
  }
  asm volatile("s_wait_asynccnt 0x0" ::: "memory");
  __syncthreads();

  const int row0 = blockIdx.x * 64 + wave * 16;
  if (row0 >= M) return;  // whole-wave uniform -> EXEC all-ones for WMMA

  const int nr = lane & 15;          // A row / B,C col within 16-wide subtile
  const int kh = (lane >> 4) << 1;   // K sub-offset: 0 or 2

  const float* __restrict__ arow = A + (size_t)(row0 + nr) * K;

  v8f acc0 = {}, acc1 = {}, acc2 = {}, acc3 = {};
  for (int k0 = 0; k0 < K; k0 += 4) {
    const v2f a = *(const v2f*)(arow + k0 + kh);  // global_load_b64, reused 4x
    const float* bk0 = &ldsB[(k0 + kh) << 6];
    const float* bk1 = &ldsB[(k0 + kh + 1) << 6];
    WMMA_STEP(acc0, 0);
    WMMA_STEP(acc1, 16);
    WMMA_STEP(acc2, 32);
    WMMA_STEP(acc3, 48);
  }

  const int rbase = row0 + ((lane >> 4) << 3);
  float* __restrict__ cbase = C + (size_t)rbase * Nc + col0 + nr;
#pragma unroll
  for (int v = 0; v < 8; ++v) {
    float* crow = cbase + (size_t)v * Nc;
    crow[0] = acc0[v];
    crow[16] = acc1[v];
    crow[32] = acc2[v];
    crow[48] = acc3[v];
  }
}

// ---------------------------------------------------------------------------
// Per-node attention logits: alpha_s[n] = xw[n,:].a_s ; alpha_d[n] = xw[n,:].a_d
// ---------------------------------------------------------------------------
__global__ __launch_bounds__(256)
void node_alpha(const float* __restrict__ xw, const float* __restrict__ a_s,
                const float* __restrict__ a_d, float* __restrict__ alpha_s,
                float* __restrict__ alpha_d, int N, int F) {
  const int wave = threadIdx.x >> 5;
  const int lane = threadIdx.x & 31;
  const int n = blockIdx.x * 8 + wave;
  if (n >= N) return;
  const float* __restrict__ row = xw + (size_t)n * F;
  float ss = 0.f, sd = 0.f;
  for (int f = lane; f < F; f += 32) {
    float x = row[f];
    ss += x * a_s[f];
    sd += x * a_d[f];
  }
#pragma unroll
  for (int off = 16; off > 0; off >>= 1) {
    ss += __shfl_xor(ss, off, 32);
    sd += __shfl_xor(sd, off, 32);
  }
  if (lane == 0) { alpha_s[n] = ss; alpha_d[n] = sd; }
}

__global__ __launch_bounds__(256)
void init_softmax_state(float* __restrict__ m, float* __restrict__ denom, int N) {
  int i = blockIdx.x * blockDim.x + threadIdx.x;
  if (i < N) { m[i] = -INFINITY; denom[i] = 0.f; }
}

__device__ __forceinline__ long long edge_src(const long long* ei, long long e,
                                              long long E) {
  return (e < E) ? ei[e] : (e - E);
}
__device__ __forceinline__ long long edge_dst(const long long* ei, long long e,
                                              long long E) {
  return (e < E) ? ei[E + e] : (e - E);
}
__device__ __forceinline__ float edge_logit(const float* as, const float* ad,
                                            long long s, long long d) {
  float v = as[s] + ad[d];
  return v >= 0.f ? v : NEG_SLOPE * v;
}
// float atomic max via signed/unsigned bit-pattern trick (init must be -inf)
__device__ __forceinline__ void atomicMaxF(float* addr, float v) {
  if (v >= 0.f) atomicMax((int*)addr, __float_as_int(v));
  else          atomicMin((unsigned int*)addr, __float_as_uint(v));
}

__global__ __launch_bounds__(256)
void edge_max(const long long* __restrict__ ei, long long E, long long NE,
              const float* __restrict__ as, const float* __restrict__ ad,
              float* __restrict__ m) {
  long long e = (long long)blockIdx.x * blockDim.x + threadIdx.x;
  if (e >= NE) return;
  long long s = edge_src(ei, e, E), d = edge_dst(ei, e, E);
  atomicMaxF(&m[d], edge_logit(as, ad, s, d));
}

__global__ __launch_bounds__(256)
void edge_expsum(const long long* __restrict__ ei, long long E, long long NE,
                 const float* __restrict__ as, const float* __restrict__ ad,
                 const float* __restrict__ m, float* __restrict__ denom) {
  long long e = (long long)blockIdx.x * blockDim.x + threadIdx.x;
  if (e >= NE) return;
  long long s = edge_src(ei, e, E), d = edge_dst(ei, e, E);
  atomicAdd(&denom[d], __expf(edge_logit(as, ad, s, d) - m[d]));
}

// out[d,:] += att * xw[s,:]  (one wave32 per edge, float4 gather)
__global__ __launch_bounds__(256)
void edge_aggregate(const long long* __restrict__ ei, long long E, long long NE,
                    const float* __restrict__ as, const float* __restrict__ ad,
                    const float* __restrict__ m, const float* __restrict__ denom,
                    const float* __restrict__ xw, float* __restrict__ out, int F) {
  const int wave = threadIdx.x >> 5;
  const int lane = threadIdx.x & 31;
  long long e = (long long)blockIdx.x * 8 + wave;
  if (e >= NE) return;
  long long s = edge_src(ei, e, E), d = edge_dst(ei, e, E);
  const float4* __restrict__ xs = (const float4*)(xw + (size_t)s * F);
  __builtin_prefetch(xs + lane, 0, 0);  // global_prefetch_b8 of gathered row
  float att = __expf(edge_logit(as, ad, s, d) - m[d]) / denom[d];
  float* __restrict__ od = out + (size_t)d * F;
  const int nchunk = F >> 2;
  for (int c = lane; c < nchunk; c += 32) {
    float4 x4 = xs[c];
    atomicAdd(od + 4 * c + 0, att * x4.x);
    atomicAdd(od + 4 * c + 1, att * x4.y);
    atomicAdd(od + 4 * c + 2, att * x4.z);
    atomicAdd(od + 4 * c + 3, att * x4.w);
  }
}

__global__ __launch_bounds__(256)
void bias_act(float* __restrict__ y, const float* __restrict__ b, long long total,
              int F, int do_relu) {
  long long i = (long long)blockIdx.x * blockDim.x + threadIdx.x;
  if (i >= total) return;
  float v = y[i] + b[i % F];
  y[i] = do_relu ? fmaxf(v, 0.f) : v;
}

// ---------------------------------------------------------------------------
static inline size_t align256(size_t x) { return (x + 255) & ~(size_t)255; }

static void run_layer(const float* x_in, const long long* ei, long long E,
                      const float* W, const float* a_s, const float* a_d,
                      const float* b, int N, int K, int F, float* xw,
                      float* acc_out, float* alpha_s, float* alpha_d, float* m,
                      float* denom, int do_relu, hipStream_t stream) {
  const long long NE = E + N;
  // GEMM: xw = x_in * W   (64x64 tile per block, B panel in LDS)
  dim3 ggrd((N + 63) / 64, F / 64);
  size_t lds_bytes = (size_t)K * 64 * sizeof(float);
  gemm_wmma_f32<<<ggrd, 128, lds_bytes, stream>>>(x_in, W, xw, N, K, F);
  // per-node logits
  node_alpha<<<(N + 7) / 8, 256, 0, stream>>>(xw, a_s, a_d, alpha_s, alpha_d, N, F);
  // edge softmax
  init_softmax_state<<<(N + 255) / 256, 256, 0, stream>>>(m, denom, N);
  int eblocks = (int)((NE + 255) / 256);
  edge_max<<<eblocks, 256, 0, stream>>>(ei, E, NE, alpha_s, alpha_d, m);
  edge_expsum<<<eblocks, 256, 0, stream>>>(ei, E, NE, alpha_s, alpha_d, m, denom);
  // weighted aggregation into acc_out (pre-zeroed by caller)
  int ablocks = (int)((NE + 7) / 8);
  edge_aggregate<<<ablocks, 256, 0, stream>>>(ei, E, NE, alpha_s, alpha_d, m,
                                              denom, xw, acc_out, F);
  long long total = (long long)N * F;
  bias_act<<<(int)((total + 255) / 256), 256, 0, stream>>>(acc_out, b, total, F,
                                                           do_relu);
}

extern "C" void kernel_launch(void* const* d_in, const int* in_sizes, int n_in,
                              void* d_out, int out_size, void* d_ws, size_t ws_size,
                              hipStream_t stream) {
  const float*     x   = (const float*)d_in[0];
  const long long* ei  = (const long long*)d_in[1];   // int64 edge_index [2,E]
  const float*     W1  = (const float*)d_in[2];
  const float*     a1s = (const float*)d_in[3];
  const float*     a1d = (const float*)d_in[4];
  const float*     b1  = (const float*)d_in[5];
  const float*     W2  = (const float*)d_in[6];
  const float*     a2s = (const float*)d_in[7];
  const float*     a2d = (const float*)d_in[8];
  const float*     b2  = (const float*)d_in[9];

  const int H    = in_sizes[3];             // 256
  const int Fout = in_sizes[7];             // 128
  const int Fin  = in_sizes[2] / H;         // 128
  const int N    = in_sizes[0] / Fin;       // 50000
  const long long E = in_sizes[1] / 2;      // 800000

  char* ws = (char*)d_ws;
  size_t off = 0;
  float* xw  = (float*)(ws + off); off += align256((size_t)N * H * sizeof(float));
  float* acc = (float*)(ws + off); off += align256((size_t)N * H * sizeof(float));
  float* alpha_s = (float*)(ws + off); off += align256((size_t)N * sizeof(float));
  float* alpha_d = (float*)(ws + off); off += align256((size_t)N * sizeof(float));
  float* m       = (float*)(ws + off); off += align256((size_t)N * sizeof(float));
  float* denom   = (float*)(ws + off); off += align256((size_t)N * sizeof(float));
  (void)ws_size; (void)n_in; (void)out_size;

  float* out = (float*)d_out;

  // ---- layer 1: h = relu(GAT(x; W1,a1,b1)) -> acc
  (void)hipMemsetAsync(acc, 0, (size_t)N * H * sizeof(float), stream);
  run_layer(x, ei, E, W1, a1s, a1d, b1, N, Fin, H, xw, acc, alpha_s, alpha_d, m,
            denom, /*relu=*/1, stream);

  // ---- layer 2: out = GAT(h; W2,a2,b2) -> d_out
  (void)hipMemsetAsync(out, 0, (size_t)N * Fout * sizeof(float), stream);
  run_layer(acc, ei, E, W2, a2s, a2d, b2, N, H, Fout, xw, out, alpha_s, alpha_d,
            m, denom, /*relu=*/0, stream);
}